// Density_prop_CNN_39779987095994
// MI455X (gfx1250) — compile-verified
//
#include <hip/hip_runtime.h>
#include <math.h>

typedef __attribute__((ext_vector_type(2))) float v2f;
typedef __attribute__((ext_vector_type(8))) float v8f;

#define NB 16       // batch
#define HWIN 28
#define KK 5
#define NC 16       // channels
#define HW1 24
#define HW2 12
#define P1 576      // 24*24
#define P2 144      // 12*12
#define UNITS 10
#define RK 25       // patch length (5*5*1)
#define EPSV 1e-4f
#define DSCALE 1.25f   // 1/(1-0.2)

#define POFF(k) ((((k) / KK) * HWIN) + ((k) % KK))   // patch element offset in x

// workspace layout (float offsets)
#define WS_MU    0                    // NB*P1*NC = 147456
#define WS_MEAN  147456               // 16
#define WS_ISTD  (WS_MEAN + 16)       // 16
#define WS_KC    (WS_ISTD + 16)       // 16
#define WS_S1    (WS_KC + 16)         // NB*100 = 1600
#define WS_TR    (WS_S1 + NB*100)     // 16
#define WS_MSQ   (WS_TR + NB)         // 16
#define WS_MUO   (WS_MSQ + NB)        // NB*UNITS = 160
#define ACC_FLOATS (NB*100 + NB + NB + NB*UNITS)

__global__ __launch_bounds__(256) void k_zero(float* __restrict__ ws) {
  int t = blockIdx.x * blockDim.x + threadIdx.x;
  if (t < ACC_FLOATS) ws[WS_S1 + t] = 0.0f;
}

// ---- Conv 5x5 (1->16) as implicit GEMM [9216 x 25] x [25 x 16] via f32 WMMA ----
__global__ __launch_bounds__(32) void k_conv(const float* __restrict__ x,
                                             const float* __restrict__ wconv,
                                             float* __restrict__ ws) {
  const int lane  = threadIdx.x;
  const int Mlo   = lane & 15;
  const int khalf = lane >> 4;              // 0/1
  const int n     = lane & 15;              // output channel
  const int wbase = lane + (lane & 16);     // khalf*32 + n
  const int m0    = blockIdx.x * 16;        // row tile base in [0, 9216)
  const int m     = m0 + Mlo;
  const int bb    = m / P1;
  const int p     = m % P1;
  const int oy    = p / HW1, ox = p % HW1;
  const float* xp = x + bb * HWIN * HWIN + oy * HWIN + ox;  // patch base

  v8f acc = {};
  // main K steps: k = kb + 2*khalf + {0,1} <= 23 < 25, no predication needed
#pragma unroll
  for (int kb = 0; kb < 24; kb += 4) {
    v2f a, bf;
    a.x  = xp[khalf ? POFF(kb + 2) : POFF(kb)];
    a.y  = xp[khalf ? POFF(kb + 3) : POFF(kb + 1)];
    bf.x = wconv[kb * NC + wbase];
    bf.y = wconv[kb * NC + wbase + NC];
    acc = __builtin_amdgcn_wmma_f32_16x16x4_f32(false, a, false, bf, (short)0, acc, false, false);
  }
  // tail: only k=24 valid, and only in the khalf==0 slot (unconditional load + cndmask)
  {
    const float ax = xp[POFF(24)];
    const float bx = wconv[24 * NC + n];
    v2f a, bf;
    a.x  = khalf ? 0.0f : ax;  a.y  = 0.0f;
    bf.x = khalf ? 0.0f : bx;  bf.y = 0.0f;
    acc = __builtin_amdgcn_wmma_f32_16x16x4_f32(false, a, false, bf, (short)0, acc, false, false);
  }
#pragma unroll
  for (int v = 0; v < 8; ++v) {
    const int row = m0 + v + khalf * 8;
    ws[WS_MU + row * NC + n] = acc[v];
  }
}

// ---- per-channel batch mean/var, k_c = softplus(w_sigma_conv)/(var+eps) ----
__global__ __launch_bounds__(256) void k_stats(const float* __restrict__ wsig,
                                               float* __restrict__ ws) {
  __shared__ float ssum[16][16];
  __shared__ float ssq[16][16];
  const int c = threadIdx.x & 15;
  const int t = threadIdx.x >> 4;
  float s = 0.0f, q = 0.0f;
  for (int i = t; i < NB * P1; i += 16) {
    float v = ws[WS_MU + i * NC + c];
    s += v; q += v * v;
  }
  ssum[t][c] = s; ssq[t][c] = q;
  __syncthreads();
  if (t == 0) {
    float S = 0.0f, Q = 0.0f;
    for (int j = 0; j < 16; ++j) { S += ssum[j][c]; Q += ssq[j][c]; }
    const float N = (float)(NB * P1);
    const float mean = S / N;
    const float var  = Q / N - mean * mean;
    ws[WS_MEAN + c] = mean;
    ws[WS_ISTD + c] = rsqrtf(var + EPSV);
    ws[WS_KC   + c] = log1pf(expf(wsig[c])) / (var + EPSV);
  }
}

// ---- fused BN/ELU/maxpool/dropout -> Y (rank-25 factor) -> Z = Y^T W1 (WMMA) ----
// one wave32 per (batch, channel): grid = 256
#define YSTR 48    // Y row stride: rows kp and kp+2 land in disjoint LDS bank halves
__global__ __launch_bounds__(32) void k_main(const float* __restrict__ x,
                                             const int* __restrict__ dmask,
                                             const float* __restrict__ wfc,
                                             float* __restrict__ ws) {
  __shared__ float Ysh[P2 * YSTR];   // Y[p][r], cols 25..31 zero-padded
  __shared__ float Wl[P2 * 16];      // W1[c][p][u], cols 10..15 zero-padded
  __shared__ float Z[32 * 16];

  const int lane = threadIdx.x;
  const int bb = blockIdx.x >> 4;
  const int c  = blockIdx.x & 15;

  const float mean = ws[WS_MEAN + c];
  const float istd = ws[WS_ISTD + c];
  const float kc   = ws[WS_KC + c];
  const float* mu  = ws + WS_MU + bb * P1 * NC;
  const float* xb  = x + bb * HWIN * HWIN;
  const float* wc  = wfc + (c * P2) * UNITS;   // W1[c,p,u] = w_mu_fc[c*144+p, u]

  // stage zero-padded W1[c] into LDS (clamped index -> unconditional load + select)
  for (int t = lane; t < P2 * 16; t += 32) {
    const int p = t >> 4, u = t & 15;
    const float v = wc[p * UNITS + (u < UNITS ? u : UNITS - 1)];
    Wl[t] = (u < UNITS) ? v : 0.0f;
  }

  // fused pool/BN/ELU/dropout + Y build + mu_out/trace/|mu|^2 partials
  float msqp = 0.0f, trp = 0.0f;
  float mp[UNITS];
#pragma unroll
  for (int u = 0; u < UNITS; ++u) mp[u] = 0.0f;

  for (int p = lane; p < P2; p += 32) {
    const int i = p / HW2, j = p % HW2;
    float best = -INFINITY; int arg = 0;
#pragma unroll
    for (int dy = 0; dy < 2; ++dy)
#pragma unroll
      for (int dx = 0; dx < 2; ++dx) {
        const float v = mu[((2 * i + dy) * HW1 + (2 * j + dx)) * NC + c];
        if (v > best) { best = v; arg = dy * 2 + dx; }   // first-occurrence ties
      }
    const float mun  = (best - mean) * istd;
    const float g    = (mun > 0.0f) ? 1.0f : expf(mun);   // elu'
    const float e    = (mun > 0.0f) ? mun  : expm1f(mun); // elu
    const float keep = (float)dmask[((bb * HW2 + i) * HW2 + j) * NC + c];
    const float mud  = e * keep * DSCALE;                 // dropped mu (mu_flat value)
    const float A    = DSCALE * keep * g;                 // rank-1 row weight
    msqp += mud * mud;

    const float* wr = wfc + (p * NC + c) * UNITS;         // NHWC-flat FC row
#pragma unroll
    for (int u = 0; u < UNITS; ++u) mp[u] += mud * wr[u];

    const int pp = (2 * i + arg / 2) * HW1 + (2 * j + arg % 2);
    const float* xr = xb + (pp / HW1) * HWIN + (pp % HW1);
    float* yr = &Ysh[p * YSTR];
#pragma unroll
    for (int r = 0; r < RK; ++r) {       // POFF(r) constant-folds under unroll
      const float y = A * xr[POFF(r)];
      yr[r] = y;
      trp += y * y;
    }
#pragma unroll
    for (int r = RK; r < 32; ++r) yr[r] = 0.0f;
  }
  __syncthreads();

  // Z[r,u] = sum_p Y[p,r] * W1[c,p,u]   (M=25->32, N=10->16, K=144), f32 WMMA
  const int Mlo   = lane & 15;
  const int khalf = lane >> 4;
  const int n     = lane & 15;
  const int ybase = khalf * (2 * YSTR) + Mlo;   // row kp = kb + 2*khalf, col Mlo
  const int wbase = lane + (lane & 16);         // khalf*32 + n
  v8f acc0 = {}, acc1 = {};
#pragma unroll 4
  for (int kb = 0; kb < P2; kb += 4) {
    v2f a0, a1, bf;
    a0.x = Ysh[kb * YSTR + ybase];
    a0.y = Ysh[kb * YSTR + YSTR + ybase];
    a1.x = Ysh[kb * YSTR + ybase + 16];
    a1.y = Ysh[kb * YSTR + YSTR + ybase + 16];
    bf.x = Wl[kb * 16 + wbase];
    bf.y = Wl[kb * 16 + wbase + 16];
    acc0 = __builtin_amdgcn_wmma_f32_16x16x4_f32(false, a0, false, bf, (short)0, acc0, false, false);
    acc1 = __builtin_amdgcn_wmma_f32_16x16x4_f32(false, a1, false, bf, (short)0, acc1, false, false);
  }
#pragma unroll
  for (int v = 0; v < 8; ++v) {
    Z[(v + khalf * 8) * 16 + n]      = acc0[v];
    Z[(16 + v + khalf * 8) * 16 + n] = acc1[v];
  }
  __syncthreads();

  // S1[b,u,v] += kc * sum_r Z[r,u] Z[r,v]   (rank-25 Gram, 100 pairs)
  for (int t = lane; t < UNITS * UNITS; t += 32) {
    const int u = t / UNITS, v = t % UNITS;
    float s = 0.0f;
#pragma unroll
    for (int r = 0; r < RK; ++r) s += Z[r * 16 + u] * Z[r * 16 + v];
    atomicAdd(&ws[WS_S1 + bb * 100 + t], kc * s);
  }

  // wave32 reductions + accumulation
#pragma unroll
  for (int u = 0; u < UNITS; ++u)
    for (int off = 16; off > 0; off >>= 1) mp[u] += __shfl_down(mp[u], off);
  for (int off = 16; off > 0; off >>= 1) {
    trp  += __shfl_down(trp, off);
    msqp += __shfl_down(msqp, off);
  }
  if (lane == 0) {
    atomicAdd(&ws[WS_TR + bb], kc * trp);
    atomicAdd(&ws[WS_MSQ + bb], msqp);
#pragma unroll
    for (int u = 0; u < UNITS; ++u) atomicAdd(&ws[WS_MUO + bb * UNITS + u], mp[u]);
  }
}

// ---- finalize: Sigma_out = S1 + softplus(w_sigma_fc)*(tr + |mu|^2) * I ----
__global__ __launch_bounds__(128) void k_final(const float* __restrict__ wsf,
                                               const float* __restrict__ ws,
                                               float* __restrict__ out) {
  const int bb = blockIdx.x;
  const int t = threadIdx.x;
  const float tail = ws[WS_TR + bb] + ws[WS_MSQ + bb];
  if (t < UNITS) out[bb * UNITS + t] = ws[WS_MUO + bb * UNITS + t];
  if (t < UNITS * UNITS) {
    const int u = t / UNITS, v = t % UNITS;
    float s = ws[WS_S1 + bb * 100 + t];
    if (u == v) s += log1pf(expf(wsf[u])) * tail;
    out[NB * UNITS + bb * 100 + t] = s;
  }
}

extern "C" void kernel_launch(void* const* d_in, const int* in_sizes, int n_in,
                              void* d_out, int out_size, void* d_ws, size_t ws_size,
                              hipStream_t stream) {
  const float* x      = (const float*)d_in[0];   // [16,28,28,1]
  const int*   dmask  = (const int*)  d_in[1];   // [16,12,12,16] bool->int
  const float* wconv  = (const float*)d_in[2];   // [5,5,1,16]
  const float* wsconv = (const float*)d_in[3];   // [16]
  const float* wfc    = (const float*)d_in[4];   // [2304,10]
  const float* wsfc   = (const float*)d_in[5];   // [10]
  float* out = (float*)d_out;                    // 160 + 1600
  float* ws  = (float*)d_ws;

  k_zero <<<(ACC_FLOATS + 255) / 256, 256, 0, stream>>>(ws);
  k_conv <<<(NB * P1) / 16, 32, 0, stream>>>(x, wconv, ws);
  k_stats<<<1, 256, 0, stream>>>(wsconv, ws);
  k_main <<<NB * NC, 32, 0, stream>>>(x, dmask, wfc, ws);
  k_final<<<NB, 128, 0, stream>>>(wsfc, ws, out);
}